// DETM_74594991997317
// MI455X (gfx1250) — compile-verified
//
#include <hip/hip_runtime.h>
#include <math.h>

#define V_  30000
#define K_  50
#define E_  300
#define T_  60
#define B_  128
#define TH_ 800
#define H_  200
#define L_  3
#define DELTA_ 0.005f

#define S_    15            // V slices for beta phase
#define CHPS  125           // 16-wide chunks per slice (15*125*16 = 30000)

typedef __attribute__((ext_vector_type(2))) float v2f;
typedef __attribute__((ext_vector_type(8))) float v8f;

// workspace layout (float offsets, all 16-aligned)
#define OFF_ALPHAS 0          // T*K*E = 900000
#define OFF_ETAS   900000     // T*K   = 3000
#define OFF_INPMAP 903008     // T*H   = 12000
#define OFF_LSTM   915008     // T*H   = 12000
#define OFF_H1     927008     // B*TH  = 102400
#define OFF_H2     1029408    // B*TH  = 102400
#define OFF_MU     1131808    // B*K   = 6400
#define OFF_LS     1138208    // B*K   = 6400
#define OFF_THETA  1144608    // B*K   = 6400
#define OFF_ACC    1151008    // [0]=kl_alpha [1]=kl_eta [2]=kl_theta_sum
#define OFF_NLLB   1151024    // B = 128
#define OFF_SMAX   1151152    // T*S*64 = 57600
#define OFF_SSUM   1208752    // 57600
#define OFF_RMAX   1266352    // T*64 = 3840
#define OFF_RSUM   1270192    // 3840   (total ~5.1 MB)

// ---------------- alphas + kl_alpha ----------------
__global__ void alpha_kernel(const float* __restrict__ muq, const float* __restrict__ lsq,
                             const float* __restrict__ eps_a, float* __restrict__ alphas,
                             float* __restrict__ acc) {
  int tk = blockIdx.x, t = tk / K_, k = tk % K_, tid = threadIdx.x;
  const float logdelta = logf(DELTA_);
  float part = 0.f;
  for (int e = tid; e < E_; e += blockDim.x) {
    float mu = muq[(k * T_ + t) * E_ + e];
    float ls = lsq[(k * T_ + t) * E_ + e];
    float a  = mu + eps_a[(t * K_ + k) * E_ + e] * expf(0.5f * ls);
    alphas[(t * K_ + k) * E_ + e] = a;
    if (t == 0) {
      part += (expf(ls) + mu * mu) / (1.f + 1e-6f) - 1.f - ls;
    } else {
      float mup = muq[(k * T_ + (t - 1)) * E_ + e];
      float lsp = lsq[(k * T_ + (t - 1)) * E_ + e];
      float ap  = mup + eps_a[((t - 1) * K_ + k) * E_ + e] * expf(0.5f * lsp);
      float d = mu - ap;
      part += (expf(ls) + d * d) / (DELTA_ + 1e-6f) - 1.f + logdelta - ls;
    }
  }
  __shared__ float red[64];
  red[tid] = part; __syncthreads();
  for (int s = 32; s > 0; s >>= 1) { if (tid < s) red[tid] += red[tid + s]; __syncthreads(); }
  if (tid == 0) atomicAdd(&acc[0], 0.5f * red[0]);
}

// ---------------- inp_map = rnn_inp @ Wmap.T + bmap ----------------
__global__ void inpmap_kernel(const float* __restrict__ rnn, const float* __restrict__ Wmap,
                              const float* __restrict__ bmap, float* __restrict__ inpmap) {
  int h = blockIdx.x, t = blockIdx.y, tid = threadIdx.x;
  const float* xr = rnn + (size_t)t * V_;
  const float* wr = Wmap + (size_t)h * V_;
  float part = 0.f;
  for (int v = tid; v < V_; v += blockDim.x) part += xr[v] * wr[v];
  __shared__ float red[128];
  red[tid] = part; __syncthreads();
  for (int s = 64; s > 0; s >>= 1) { if (tid < s) red[tid] += red[tid + s]; __syncthreads(); }
  if (tid == 0) inpmap[t * H_ + h] = red[0] + bmap[h];
}

// ---------------- 3-layer LSTM + eta chain (single block, sequential) ----------------
__global__ void lstm_eta_kernel(const float* __restrict__ inpmap, const float* __restrict__ Wih,
                                const float* __restrict__ Whh, const float* __restrict__ bih,
                                const float* __restrict__ bhh, const float* __restrict__ Wmu,
                                const float* __restrict__ bmu, const float* __restrict__ Wls,
                                const float* __restrict__ bls, const float* __restrict__ eps_e,
                                float* __restrict__ lstmbuf, float* __restrict__ etas,
                                float* __restrict__ acc) {
  __shared__ float x[H_], hs[H_], cs[H_], g[4 * H_];
  __shared__ float inp[H_ + K_], eta[K_], red[64];
  int tid = threadIdx.x;                      // 256 threads
  for (int l = 0; l < L_; ++l) {
    for (int i = tid; i < H_; i += 256) { hs[i] = 0.f; cs[i] = 0.f; }
    __syncthreads();
    for (int t = 0; t < T_; ++t) {
      const float* src = (l == 0) ? (inpmap + t * H_) : (lstmbuf + t * H_);
      for (int i = tid; i < H_; i += 256) x[i] = src[i];
      __syncthreads();
      for (int j = tid; j < 4 * H_; j += 256) {
        float s = bih[l * 4 * H_ + j] + bhh[l * 4 * H_ + j];
        const float* wi = Wih + (size_t)(l * 4 * H_ + j) * H_;
        const float* wh = Whh + (size_t)(l * 4 * H_ + j) * H_;
        for (int i = 0; i < H_; ++i) s += wi[i] * x[i] + wh[i] * hs[i];
        g[j] = s;
      }
      __syncthreads();
      for (int i = tid; i < H_; i += 256) {
        float ii = g[i], ff = g[H_ + i], gg = g[2 * H_ + i], oo = g[3 * H_ + i];
        float c = (1.f / (1.f + expf(-ff))) * cs[i] + (1.f / (1.f + expf(-ii))) * tanhf(gg);
        float h = (1.f / (1.f + expf(-oo))) * tanhf(c);
        cs[i] = c; hs[i] = h; lstmbuf[t * H_ + i] = h;
      }
      __syncthreads();
    }
  }
  // eta chain
  for (int i = tid; i < K_; i += 256) eta[i] = 0.f;
  __syncthreads();
  const float logdelta = logf(DELTA_);
  float klsum = 0.f;
  for (int t = 0; t < T_; ++t) {
    for (int i = tid; i < H_; i += 256) inp[i] = lstmbuf[t * H_ + i];
    for (int i = tid; i < K_; i += 256) inp[H_ + i] = eta[i];
    __syncthreads();
    float mu = 0.f, ls = 0.f, kl = 0.f, etav = 0.f;
    if (tid < K_) {
      mu = bmu[tid]; ls = bls[tid];
      const float* wm = Wmu + (size_t)tid * (H_ + K_);
      const float* wl = Wls + (size_t)tid * (H_ + K_);
      for (int i = 0; i < H_ + K_; ++i) { mu += wm[i] * inp[i]; ls += wl[i] * inp[i]; }
      etav = mu + eps_e[t * K_ + tid] * expf(0.5f * ls);
      if (t == 0) {
        kl = (expf(ls) + mu * mu) / (1.f + 1e-6f) - 1.f - ls;
      } else {
        float d = mu - inp[H_ + tid];
        kl = (expf(ls) + d * d) / (DELTA_ + 1e-6f) - 1.f + logdelta - ls;
      }
      etas[t * K_ + tid] = etav;
    }
    if (tid < 64) red[tid] = (tid < K_) ? kl : 0.f;
    __syncthreads();
    if (tid < K_) eta[tid] = etav;
    for (int s = 32; s > 0; s >>= 1) { if (tid < s) red[tid] += red[tid + s]; __syncthreads(); }
    if (tid == 0) klsum += red[0];
    __syncthreads();
  }
  if (tid == 0) acc[1] = 0.5f * klsum;
}

// ---------------- h1 = relu(concat(nb, eta_td) @ W1.T + b1), fp32 WMMA ----------------
// Guarded lanes load from a clamped valid row (unconditionally); garbage results land in
// columns that the store guard drops, so no divergent loads in the hot loop.
__global__ void __launch_bounds__(128) qt_gemm_kernel(const float* __restrict__ nb,
    const int* __restrict__ times, const float* __restrict__ etas,
    const float* __restrict__ W1, const float* __restrict__ b1, float* __restrict__ h1) {
  int w = threadIdx.x >> 5, L = threadIdx.x & 31;
  int nbase = blockIdx.x * 64 + w * 16;
  int mbase = blockIdx.y * 16;
  if (nbase >= TH_) return;                 // wave-uniform
  int g = L >> 4, lm = L & 15, eoff = g * 2;
  int mrow = mbase + lm;                    // always < 128
  int nrow = nbase + lm;
  int nclamp = (nrow < TH_) ? nrow : 0;
  int tm = times[mrow];
  const float* arow = nb + (size_t)mrow * V_;
  const float* brow = W1 + (size_t)nclamp * (V_ + K_);
  v8f c = {};
  for (int e0 = 0; e0 < V_; e0 += 4) {
    v2f a = *reinterpret_cast<const v2f*>(arow + e0 + eoff);
    v2f b = *reinterpret_cast<const v2f*>(brow + e0 + eoff);
    c = __builtin_amdgcn_wmma_f32_16x16x4_f32(false, a, false, b, (short)0, c, false, false);
  }
  for (int e0 = V_; e0 < V_ + 52; e0 += 4) {     // 13 tail iterations
    int i0 = e0 + eoff - V_;
    v2f a = {0.f, 0.f}, b = {0.f, 0.f};
    if (i0 < K_)     a.x = etas[tm * K_ + i0];
    if (i0 + 1 < K_) a.y = etas[tm * K_ + i0 + 1];
    if (i0 < K_)     b.x = W1[(size_t)nclamp * (V_ + K_) + V_ + i0];
    if (i0 + 1 < K_) b.y = W1[(size_t)nclamp * (V_ + K_) + V_ + i0 + 1];
    c = __builtin_amdgcn_wmma_f32_16x16x4_f32(false, a, false, b, (short)0, c, false, false);
  }
#pragma unroll
  for (int r = 0; r < 8; ++r) {
    int row = mbase + r + g * 8, col = nbase + lm;
    if (col < TH_) h1[(size_t)row * TH_ + col] = fmaxf(c[r] + b1[col], 0.f);
  }
}

// ---------------- generic fp32 WMMA NT-GEMM: C = A*B^T (+bias, relu) ----------------
// Requires: Kd % 4 == 0, lda/ldb even (true for all call sites: Kd=800).
__global__ void __launch_bounds__(128) gemm_nt_kernel(const float* __restrict__ A,
    const float* __restrict__ B, const float* __restrict__ bias, float* __restrict__ C,
    int M, int N, int Kd, int lda, int ldb, int ldc, int relu) {
  int w = threadIdx.x >> 5, L = threadIdx.x & 31;
  int nbase = blockIdx.x * 64 + w * 16;
  int mbase = blockIdx.y * 16;
  if (nbase >= N) return;                   // wave-uniform
  int g = L >> 4, lm = L & 15, eoff = g * 2;
  int mrow = mbase + lm, nrow = nbase + lm;
  int mclamp = (mrow < M) ? mrow : 0;
  int nclamp = (nrow < N) ? nrow : 0;
  const float* arow = A + (size_t)mclamp * lda;
  const float* brow = B + (size_t)nclamp * ldb;
  v8f c = {};
  for (int e0 = 0; e0 < Kd; e0 += 4) {
    v2f a = *reinterpret_cast<const v2f*>(arow + e0 + eoff);
    v2f b = *reinterpret_cast<const v2f*>(brow + e0 + eoff);
    c = __builtin_amdgcn_wmma_f32_16x16x4_f32(false, a, false, b, (short)0, c, false, false);
  }
#pragma unroll
  for (int r = 0; r < 8; ++r) {
    int row = mbase + r + g * 8, col = nbase + lm;
    if (row < M && col < N) {
      float v = c[r] + bias[col];
      if (relu) v = fmaxf(v, 0.f);
      C[(size_t)row * ldc + col] = v;
    }
  }
}

// ---------------- theta = softmax(z), kl_theta ----------------
__global__ void theta_kernel(const float* __restrict__ mu, const float* __restrict__ lsv,
                             const float* __restrict__ eps_t, const int* __restrict__ times,
                             const float* __restrict__ etas, float* __restrict__ theta,
                             float* __restrict__ acc) {
  int b = blockIdx.x, tid = threadIdx.x;    // 64 threads
  __shared__ float red[64];
  float m = 0.f, l = 0.f, z = -__builtin_inff();
  if (tid < K_) {
    m = mu[b * K_ + tid]; l = lsv[b * K_ + tid];
    z = m + eps_t[b * K_ + tid] * expf(0.5f * l);
  }
  red[tid] = z; __syncthreads();
  for (int s = 32; s > 0; s >>= 1) { if (tid < s) red[tid] = fmaxf(red[tid], red[tid + s]); __syncthreads(); }
  float zmax = red[0]; __syncthreads();
  float ez = (tid < K_) ? expf(z - zmax) : 0.f;
  red[tid] = ez; __syncthreads();
  for (int s = 32; s > 0; s >>= 1) { if (tid < s) red[tid] += red[tid + s]; __syncthreads(); }
  float zs = red[0]; __syncthreads();
  if (tid < K_) theta[b * K_ + tid] = ez / zs;
  float kl = 0.f;
  if (tid < K_) {
    float d = m - etas[times[b] * K_ + tid];
    kl = (expf(l) + d * d) / (1.f + 1e-6f) - 1.f - l;
  }
  red[tid] = kl; __syncthreads();
  for (int s = 32; s > 0; s >>= 1) { if (tid < s) red[tid] += red[tid + s]; __syncthreads(); }
  if (tid == 0) atomicAdd(&acc[2], 0.5f * red[0]);
}

// ---------------- beta phase 1: per-(t, slice) online-softmax stats ----------------
__global__ void __launch_bounds__(128) beta_stats_kernel(const float* __restrict__ alphas,
    const float* __restrict__ rho, float* __restrict__ smax, float* __restrict__ ssum) {
  int t = blockIdx.x, s = blockIdx.y;
  int tid = threadIdx.x, w = tid >> 5, L = tid & 31, g = L >> 4, lm = L & 15;
  int eoff = g * 2;
  __shared__ float lmax[64][16], lsum[64][16];
  const float* At = alphas + (size_t)t * K_ * E_;
  int row = 16 * w + lm;
  const float* arow = At + (size_t)(row < K_ ? row : 0) * E_;   // clamped; padded rows dup row 0
  float rmax[8], rsum[8];
#pragma unroll
  for (int r = 0; r < 8; ++r) { rmax[r] = -__builtin_inff(); rsum[r] = 0.f; }
  for (int ch = s * CHPS; ch < (s + 1) * CHPS; ++ch) {
    int v0 = ch * 16;
    const float* brow = rho + (size_t)(v0 + lm) * E_;
    v8f c = {};
    for (int e0 = 0; e0 < E_; e0 += 4) {
      v2f a = *reinterpret_cast<const v2f*>(arow + e0 + eoff);
      v2f b = *reinterpret_cast<const v2f*>(brow + e0 + eoff);
      c = __builtin_amdgcn_wmma_f32_16x16x4_f32(false, a, false, b, (short)0, c, false, false);
    }
#pragma unroll
    for (int r = 0; r < 8; ++r) {
      float l = c[r];
      if (l > rmax[r]) { rsum[r] = rsum[r] * expf(rmax[r] - l) + 1.f; rmax[r] = l; }
      else             { rsum[r] += expf(l - rmax[r]); }
    }
  }
#pragma unroll
  for (int r = 0; r < 8; ++r) {
    int rr = 16 * w + r + g * 8;
    lmax[rr][lm] = rmax[r]; lsum[rr][lm] = rsum[r];
  }
  __syncthreads();
  if (tid < 64) {
    float m = -__builtin_inff();
    for (int i = 0; i < 16; ++i) m = fmaxf(m, lmax[tid][i]);
    float su = 0.f;
    for (int i = 0; i < 16; ++i) su += lsum[tid][i] * expf(lmax[tid][i] - m);
    smax[((size_t)t * S_ + s) * 64 + tid] = m;
    ssum[((size_t)t * S_ + s) * 64 + tid] = su;
  }
}

// ---------------- beta phase 2: combine slice stats per (t,row) ----------------
__global__ void beta_combine_kernel(const float* __restrict__ smax, const float* __restrict__ ssum,
                                    float* __restrict__ rmaxG, float* __restrict__ rsumG) {
  int t = blockIdx.x, row = threadIdx.x;    // 64 threads
  float m = -__builtin_inff();
  for (int s = 0; s < S_; ++s) m = fmaxf(m, smax[((size_t)t * S_ + s) * 64 + row]);
  float su = 0.f;
  for (int s = 0; s < S_; ++s)
    su += ssum[((size_t)t * S_ + s) * 64 + row] * expf(smax[((size_t)t * S_ + s) * 64 + row] - m);
  rmaxG[t * 64 + row] = m;
  rsumG[t * 64 + row] = su;
}

// ---------------- beta phase 3: recompute logits, beta, theta-mix, nll ----------------
__global__ void __launch_bounds__(128) beta_nll_kernel(const float* __restrict__ alphas,
    const float* __restrict__ rho, const float* __restrict__ theta,
    const float* __restrict__ bows, const int* __restrict__ times,
    const float* __restrict__ rmaxG, const float* __restrict__ rsumG,
    float* __restrict__ nllb) {
  int t = blockIdx.x, s = blockIdx.y;
  int tid = threadIdx.x, w = tid >> 5, L = tid & 31, g = L >> 4, lm = L & 15;
  int eoff = g * 2;
  __shared__ float rowmax[64], rowsum[64];
  __shared__ float betaT[4][64][17];
  __shared__ int timesS[B_];
  if (tid < 64) { rowmax[tid] = rmaxG[t * 64 + tid]; rowsum[tid] = rsumG[t * 64 + tid]; }
  if (tid < B_) timesS[tid] = times[tid];
  __syncthreads();
  const float* At = alphas + (size_t)t * K_ * E_;
  const int nj = (CHPS + 3) / 4;            // 32
  for (int j = 0; j < nj; ++j) {
    int chl = j * 4 + w;
    bool active = chl < CHPS;
    int v0 = (s * CHPS + chl) * 16;
    if (active) {
      const float* brow = rho + (size_t)(v0 + lm) * E_;
#pragma unroll
      for (int mt = 0; mt < 4; ++mt) {
        int row = 16 * mt + lm;
        const float* arow = At + (size_t)(row < K_ ? row : 0) * E_;
        v8f c = {};
        for (int e0 = 0; e0 < E_; e0 += 4) {
          v2f a = *reinterpret_cast<const v2f*>(arow + e0 + eoff);
          v2f b = *reinterpret_cast<const v2f*>(brow + e0 + eoff);
          c = __builtin_amdgcn_wmma_f32_16x16x4_f32(false, a, false, b, (short)0, c, false, false);
        }
#pragma unroll
        for (int r = 0; r < 8; ++r) {
          int kr = 16 * mt + r + g * 8;
          betaT[w][kr][lm] = expf(c[r] - rowmax[kr]) / rowsum[kr];
        }
      }
    }
    __syncthreads();                        // uniform barrier: intra-wave LDS visibility
    if (active) {
      for (int b = 0; b < B_; ++b) {
        if (timesS[b] != t) continue;
        float part = 0.f;
        int k0 = g * 25;
        for (int k = k0; k < k0 + 25; ++k) part += theta[b * K_ + k] * betaT[w][k][lm];
        float tot = part + __shfl_down(part, 16);
        float contrib = 0.f;
        if (L < 16) contrib = -logf(tot + 1e-6f) * bows[(size_t)b * V_ + v0 + L];
        contrib += __shfl_down(contrib, 8);
        contrib += __shfl_down(contrib, 4);
        contrib += __shfl_down(contrib, 2);
        contrib += __shfl_down(contrib, 1);
        if (L == 0) atomicAdd(&nllb[b], contrib);
      }
    }
    __syncthreads();
  }
}

// ---------------- final reduction ----------------
__global__ void finalize_kernel(const float* __restrict__ acc, const float* __restrict__ nllb,
                                const int* __restrict__ ndocs, float* __restrict__ out) {
  __shared__ float red[128];
  int tid = threadIdx.x;
  red[tid] = (tid < B_) ? nllb[tid] : 0.f;
  __syncthreads();
  for (int s = 64; s > 0; s >>= 1) { if (tid < s) red[tid] += red[tid + s]; __syncthreads(); }
  if (tid == 0) {
    float coeff = (float)ndocs[0] / (float)B_;
    float nll_tot = red[0] * coeff;
    float kl_a = acc[0], kl_e = acc[1], kl_t = acc[2] * coeff;
    out[0] = nll_tot + kl_a + kl_e + kl_t;
    out[1] = nll_tot; out[2] = kl_a; out[3] = kl_e; out[4] = kl_t;
  }
}

extern "C" void kernel_launch(void* const* d_in, const int* in_sizes, int n_in,
                              void* d_out, int out_size, void* d_ws, size_t ws_size,
                              hipStream_t stream) {
  (void)in_sizes; (void)n_in; (void)out_size; (void)ws_size;
  const float* bows  = (const float*)d_in[0];
  const float* nb    = (const float*)d_in[1];
  const int*   times = (const int*)d_in[2];
  const float* rnn   = (const float*)d_in[3];
  const int*   ndocs = (const int*)d_in[4];
  const float* eps_a = (const float*)d_in[5];
  const float* eps_e = (const float*)d_in[6];
  const float* eps_t = (const float*)d_in[7];
  const float* rho   = (const float*)d_in[8];
  const float* muqa  = (const float*)d_in[9];
  const float* lsqa  = (const float*)d_in[10];
  const float* W1    = (const float*)d_in[11];
  const float* b1    = (const float*)d_in[12];
  const float* W2    = (const float*)d_in[13];
  const float* b2    = (const float*)d_in[14];
  const float* Wmu_t = (const float*)d_in[15];
  const float* bmu_t = (const float*)d_in[16];
  const float* Wls_t = (const float*)d_in[17];
  const float* bls_t = (const float*)d_in[18];
  const float* Wmap  = (const float*)d_in[19];
  const float* bmap  = (const float*)d_in[20];
  const float* Wih   = (const float*)d_in[21];
  const float* Whh   = (const float*)d_in[22];
  const float* bih   = (const float*)d_in[23];
  const float* bhh   = (const float*)d_in[24];
  const float* Wmu_e = (const float*)d_in[25];
  const float* bmu_e = (const float*)d_in[26];
  const float* Wls_e = (const float*)d_in[27];
  const float* bls_e = (const float*)d_in[28];

  float* ws      = (float*)d_ws;
  float* alphas  = ws + OFF_ALPHAS;
  float* etas    = ws + OFF_ETAS;
  float* inpmap  = ws + OFF_INPMAP;
  float* lstmbuf = ws + OFF_LSTM;
  float* h1      = ws + OFF_H1;
  float* h2      = ws + OFF_H2;
  float* muth    = ws + OFF_MU;
  float* lsth    = ws + OFF_LS;
  float* theta   = ws + OFF_THETA;
  float* acc     = ws + OFF_ACC;
  float* nllb    = ws + OFF_NLLB;
  float* smax    = ws + OFF_SMAX;
  float* ssum    = ws + OFF_SSUM;
  float* rmaxG   = ws + OFF_RMAX;
  float* rsumG   = ws + OFF_RSUM;

  hipMemsetAsync(acc, 0, (16 + B_) * sizeof(float), stream);

  alpha_kernel<<<T_ * K_, 64, 0, stream>>>(muqa, lsqa, eps_a, alphas, acc);
  inpmap_kernel<<<dim3(H_, T_), 128, 0, stream>>>(rnn, Wmap, bmap, inpmap);
  lstm_eta_kernel<<<1, 256, 0, stream>>>(inpmap, Wih, Whh, bih, bhh,
                                         Wmu_e, bmu_e, Wls_e, bls_e, eps_e,
                                         lstmbuf, etas, acc);
  qt_gemm_kernel<<<dim3(13, 8), 128, 0, stream>>>(nb, times, etas, W1, b1, h1);
  gemm_nt_kernel<<<dim3(13, 8), 128, 0, stream>>>(h1, W2, b2, h2,
                                                  B_, TH_, TH_, TH_, TH_, TH_, 1);
  gemm_nt_kernel<<<dim3(1, 8), 128, 0, stream>>>(h2, Wmu_t, bmu_t, muth,
                                                 B_, K_, TH_, TH_, TH_, K_, 0);
  gemm_nt_kernel<<<dim3(1, 8), 128, 0, stream>>>(h2, Wls_t, bls_t, lsth,
                                                 B_, K_, TH_, TH_, TH_, K_, 0);
  theta_kernel<<<B_, 64, 0, stream>>>(muth, lsth, eps_t, times, etas, theta, acc);
  beta_stats_kernel<<<dim3(T_, S_), 128, 0, stream>>>(alphas, rho, smax, ssum);
  beta_combine_kernel<<<T_, 64, 0, stream>>>(smax, ssum, rmaxG, rsumG);
  beta_nll_kernel<<<dim3(T_, S_), 128, 0, stream>>>(alphas, rho, theta, bows, times,
                                                    rmaxG, rsumG, nllb);
  finalize_kernel<<<1, 128, 0, stream>>>(acc, nllb, ndocs, (float*)d_out);
}